// DotProdAttention_1580547970581
// MI455X (gfx1250) — compile-verified
//
#include <hip/hip_runtime.h>
#include <math.h>

// Shapes fixed by the reference
#define B_     32
#define S_     4096
#define H_     1024
#define CHUNKS 32           // split-S factor
#define SCHUNK 128          // S_/CHUNKS, = 8 waves * 16 rows
#define TPB    256          // 8 wave32s

typedef __attribute__((ext_vector_type(2))) float v2f;
typedef __attribute__((ext_vector_type(8))) float v8f;

// ---------------------------------------------------------------------------
// Pass 1: one workgroup per (batch, S-chunk).
//   Phase A: scores via V_WMMA_F32_16X16X4_F32 (K-tile in A, q broadcast in B;
//            the matrix unit performs the k-reduction, fp32 end-to-end).
//   Phase B: chunk softmax stats + unnormalized weighted accumulation acc[H].
// K is streamed from HBM exactly once; phase-B re-read of the 512 KB chunk
// follows immediately and hits L2.
// ---------------------------------------------------------------------------
__global__ __launch_bounds__(TPB) void attn_pass1(
    const float* __restrict__ q,      // (B,1,H)
    const float* __restrict__ ctx,    // (B,S,H)
    float* __restrict__ ws_m,         // (B,CHUNKS)  chunk max
    float* __restrict__ ws_sum,       // (B,CHUNKS)  chunk sum of exp
    float* __restrict__ ws_w,         // (B,S)       exp(score - m_chunk)
    float* __restrict__ ws_acc)       // (B,CHUNKS,H) unnormalized weighted sum
{
    __shared__ float s_q[H_];
    __shared__ float s_scores[SCHUNK];
    __shared__ float s_w[SCHUNK];

    const int c    = blockIdx.x;        // S-chunk index
    const int b    = blockIdx.y;        // batch
    const int t    = threadIdx.x;
    const int wave = t >> 5;
    const int lane = t & 31;

    // stage q in LDS (4 KB), reused by all 256 WMMA B-fragments per wave
    for (int i = t; i < H_; i += TPB) s_q[i] = q[(size_t)b * H_ + i];
    __syncthreads();

    // ---------------- Phase A: scores via WMMA f32 16x16x4 ----------------
    // A-fragment layout (16x4 f32): lane L holds rows r=L%16, k = 2*(L/16)+v
    const int r  = lane & 15;
    const int kk = (lane >> 4) << 1;
    const int srow0 = c * SCHUNK + wave * 16;     // this wave's 16 s-rows
    const float* rowp = ctx + ((size_t)b * S_ + srow0 + r) * (size_t)H_ + kk;

    v8f d = {};                                    // fp32 accumulator D
    for (int h0 = 0; h0 < H_; h0 += 4) {
        const float2 av = *(const float2*)(rowp + h0);          // K[s, h0+kk..+1]
        const float2 qv = *(const float2*)&s_q[h0 + kk];        // q broadcast cols
        v2f a;  a.x = av.x;  a.y = av.y;
        v2f bq; bq.x = qv.x; bq.y = qv.y;
        // D += A(16x4 K-tile) x B(4x16 q-broadcast): D[i,j] = score(srow0+i)
        d = __builtin_amdgcn_wmma_f32_16x16x4_f32(
                /*neg_a=*/false, a, /*neg_b=*/false, bq,
                /*c_mod=*/(short)0, d, /*reuse_a=*/false, /*reuse_b=*/false);
    }
    // D columns are identical; lane 0 holds rows 0..7 in d[0..7], lane 16 rows 8..15
    if (lane == 0) {
        #pragma unroll
        for (int i = 0; i < 8; ++i) s_scores[wave * 16 + i] = d[i];
    } else if (lane == 16) {
        #pragma unroll
        for (int i = 0; i < 8; ++i) s_scores[wave * 16 + 8 + i] = d[i];
    }
    __syncthreads();

    // ------------- chunk softmax stats (tiny, redundant scan) -------------
    float m = -INFINITY;
    for (int s = 0; s < SCHUNK; ++s) m = fmaxf(m, s_scores[s]);
    float sum = 0.0f;
    for (int s = 0; s < SCHUNK; ++s) sum += __expf(s_scores[s] - m);
    if (t < SCHUNK) s_w[t] = __expf(s_scores[t] - m);
    __syncthreads();

    if (t == 0) {
        ws_m  [b * CHUNKS + c] = m;
        ws_sum[b * CHUNKS + c] = sum;
    }
    if (t < SCHUNK)
        ws_w[(size_t)b * S_ + c * SCHUNK + t] = s_w[t];

    // ------------- Phase B: acc[h] = sum_s w[s] * K[s,h] ------------------
    // thread t owns 4 contiguous columns -> fully coalesced b128 streams
    const float* cb = ctx + ((size_t)b * S_ + (size_t)c * SCHUNK) * (size_t)H_ + 4 * t;
    float ax = 0.f, ay = 0.f, az = 0.f, aw = 0.f;
    for (int s = 0; s < SCHUNK; ++s) {
        if (s + 8 < SCHUNK)
            __builtin_prefetch(cb + (size_t)(s + 8) * H_, 0, 0);  // global_prefetch_b8
        const float4 kv = *(const float4*)(cb + (size_t)s * H_);
        const float  w  = s_w[s];
        ax = fmaf(w, kv.x, ax);
        ay = fmaf(w, kv.y, ay);
        az = fmaf(w, kv.z, az);
        aw = fmaf(w, kv.w, aw);
    }
    float4 out4 = make_float4(ax, ay, az, aw);
    *(float4*)(ws_acc + ((size_t)(b * CHUNKS + c)) * H_ + 4 * t) = out4;
}

// ---------------------------------------------------------------------------
// Pass 2: one workgroup per batch. Merge the 32 chunk partials (flash-style
// rescale), tanh, and emit normalized attn. Traffic ~4.6 MB total: negligible.
// d_out layout: [ out (B*H) | attn (B*S) ]  (tuple return order)
// ---------------------------------------------------------------------------
__global__ __launch_bounds__(TPB) void attn_pass2(
    const float* __restrict__ ws_m,
    const float* __restrict__ ws_sum,
    const float* __restrict__ ws_w,
    const float* __restrict__ ws_acc,
    float* __restrict__ out)
{
    __shared__ float s_scale[CHUNKS];
    const int b = blockIdx.x;
    const int t = threadIdx.x;

    // global max + normalizer (redundant tiny scan per thread)
    float M = -INFINITY;
    for (int c = 0; c < CHUNKS; ++c) M = fmaxf(M, ws_m[b * CHUNKS + c]);
    float total = 0.0f;
    for (int c = 0; c < CHUNKS; ++c)
        total += ws_sum[b * CHUNKS + c] * __expf(ws_m[b * CHUNKS + c] - M);
    const float inv = 1.0f / total;
    if (t < CHUNKS) s_scale[t] = __expf(ws_m[b * CHUNKS + t] - M) * inv;
    __syncthreads();

    // out[h] = tanh( sum_c acc_c[h] * scale_c ), thread t owns 4 columns
    float ax = 0.f, ay = 0.f, az = 0.f, aw = 0.f;
    for (int c = 0; c < CHUNKS; ++c) {
        const float4 a = *(const float4*)(ws_acc + ((size_t)(b * CHUNKS + c)) * H_ + 4 * t);
        const float sc = s_scale[c];
        ax = fmaf(sc, a.x, ax);
        ay = fmaf(sc, a.y, ay);
        az = fmaf(sc, a.z, az);
        aw = fmaf(sc, a.w, aw);
    }
    float4 o4 = make_float4(tanhf(ax), tanhf(ay), tanhf(az), tanhf(aw));
    *(float4*)(out + (size_t)b * H_ + 4 * t) = o4;

    // attn[s] = exp(score - m_c) * exp(m_c - M) / total
    float* attn = out + (size_t)B_ * H_ + (size_t)b * S_;
    for (int s = t; s < S_; s += TPB)
        attn[s] = ws_w[(size_t)b * S_ + s] * s_scale[s / SCHUNK];
}

extern "C" void kernel_launch(void* const* d_in, const int* in_sizes, int n_in,
                              void* d_out, int out_size, void* d_ws, size_t ws_size,
                              hipStream_t stream) {
    const float* q   = (const float*)d_in[0];   // output: (B,1,H) fp32
    const float* ctx = (const float*)d_in[1];   // context: (B,S,H) fp32
    float* out = (float*)d_out;

    // workspace layout (floats): m | sum | w | acc  (~4.6 MB)
    float* w      = (float*)d_ws;
    float* ws_m   = w;
    float* ws_sum = ws_m   + (size_t)B_ * CHUNKS;
    float* ws_w   = ws_sum + (size_t)B_ * CHUNKS;
    float* ws_acc = ws_w   + (size_t)B_ * S_;

    dim3 g1(CHUNKS, B_);
    attn_pass1<<<g1, TPB, 0, stream>>>(q, ctx, ws_m, ws_sum, ws_w, ws_acc);
    attn_pass2<<<B_, TPB, 0, stream>>>(ws_m, ws_sum, ws_w, ws_acc, out);
}